// Plane_40681930227961
// MI455X (gfx1250) — compile-verified
//
#include <hip/hip_runtime.h>
#include <cstdint>

// ---------------- problem constants (match reference) ----------------
#define NUM_LEVELS   16
#define START_HASH   6            // levels 0..5 dense grid, 6..15 spatial hash
#define N_ENTRIES_C  524309u      // prime hashmap size
#define PS1_32       19349663u    // second spatial-hash prime (fits in u32)
#define N_POINTS_C   524288
#define LDS_LEVELS   3            // levels 0..2 staged in LDS
#define LDS_ENTRIES  5603         // 289 + 1089 + 4225 entries (float2 each) = 44,824 B

typedef unsigned int v4u __attribute__((ext_vector_type(4)));
typedef int          v8i __attribute__((ext_vector_type(8)));
typedef int          v4i __attribute__((ext_vector_type(4)));

// (L ^ ix) + (m - (42H mod m)) reduced into [0, m).  Inputs: L < 2^20, b in (0, m].
__device__ __forceinline__ uint32_t hash_fold(uint32_t L, uint32_t b, uint32_t ix) {
    uint32_t r = (L ^ ix) + b;                       // r < 2^20 + m < 3m
    r = (r >= N_ENTRIES_C) ? r - N_ENTRIES_C : r;
    r = (r >= N_ENTRIES_C) ? r - N_ENTRIES_C : r;
    return r;
}

__global__ __launch_bounds__(256)
void hashgrid_encode_kernel(const float2* __restrict__ x,
                            const float2* __restrict__ data,
                            float* __restrict__ out)
{
    // per-level base offsets into the table (compile-time)
    constexpr int off[NUM_LEVELS] = {
        0, 289, 1378, 5603, 22244, 88293, 351462,
        875771, 1400080, 1924389, 2448698, 2973007,
        3497316, 4021625, 4545934, 5070243
    };

    __shared__ float2 smem[LDS_ENTRIES];

    // ---- TDM: DMA levels 0..2 of the table into LDS (one wave issues) ----
    if (threadIdx.x == 0) {
        uint32_t lds_addr = (uint32_t)(uintptr_t)&smem[0];   // LDS aperture: low 32 bits = LDS byte offset
        uint64_t gaddr    = (uint64_t)(uintptr_t)data;

        // D# group 0: count=1 | lds_addr | global_addr[56:0] | type=2
        v4u g0;
        g0.x = 1u;
        g0.y = lds_addr;
        g0.z = (uint32_t)(gaddr & 0xFFFFFFFFull);
        g0.w = (uint32_t)((gaddr >> 32) & 0x01FFFFFFull) | (2u << 30);

        // D# group 1: data_size=3 (8B); tensor_dim0=5603; tensor_dim1=1;
        //             tile_dim0=5603; tile_dim1=1; strides=5603
        v8i g1;
        g1[0] = (int)(3u << 16);
        g1[1] = (int)((LDS_ENTRIES & 0xFFFF) << 16);
        g1[2] = (int)((LDS_ENTRIES >> 16) | (1u << 16));
        g1[3] = (int)(0u | ((uint32_t)LDS_ENTRIES << 16));
        g1[4] = (int)(1u);
        g1[5] = (int)LDS_ENTRIES;
        g1[6] = (int)((LDS_ENTRIES & 0xFFFF) << 16);
        g1[7] = 0;

        v4i gz4 = {0, 0, 0, 0};
        v8i gz8 = {0, 0, 0, 0, 0, 0, 0, 0};
        __builtin_amdgcn_tensor_load_to_lds(g0, g1, gz4, gz4, gz8, 0);
        __builtin_amdgcn_s_wait_tensorcnt(0);
    }
    __syncthreads();

    // ---- gather + bilinear interp, grid-stride over points ----
    const int stride = gridDim.x * blockDim.x;
    for (int p = blockIdx.x * blockDim.x + threadIdx.x; p < N_POINTS_C; p += stride) {
        const float2 xy = x[p];
        float2* __restrict__ op = (float2*)(out + (size_t)p * (2 * NUM_LEVELS));

        #pragma unroll
        for (int l = 0; l < NUM_LEVELS; ++l) {
            const int   res = 16 << l;
            const float s   = (float)res;
            const float fx = xy.x * s;
            const float fy = xy.y * s;
            // corner ints exactly as reference: trunc(fx + corner) in f32
            const int i0x = (int)fx;
            const int i0y = (int)fy;
            const int i1x = (int)(fx + 1.0f);
            const int i1y = (int)(fy + 1.0f);
            const float ox = fx - (float)i0x;
            const float oy = fy - (float)i0y;
            const float rx = 1.0f - ox, ry = 1.0f - oy;
            const float w00 = rx * ry, w01 = rx * oy, w10 = ox * ry, w11 = ox * oy;

            float2 v00, v01, v10, v11;
            if (l < START_HASH) {
                const int r1 = res + 1;
                const int a  = i0x * r1;
                const int b  = i1x * r1;
                if (l < LDS_LEVELS) {
                    const int base = off[l];
                    v00 = smem[base + a + i0y];
                    v01 = smem[base + a + i1y];
                    v10 = smem[base + b + i0y];
                    v11 = smem[base + b + i1y];
                } else {
                    const float2* __restrict__ dbase = data + off[l];
                    v00 = dbase[a + i0y];
                    v01 = dbase[a + i1y];
                    v10 = dbase[b + i0y];
                    v11 = dbase[b + i1y];
                }
            } else {
                // (ix ^ (iy*PS1)) % m, strength-reduced to 32-bit:
                //   iy*PS1 = H*2^20 + L,  2^20 ≡ -42 (mod m), ix < 2^20
                //   => hash = ((L ^ ix) + (m - (42H % m))) % m
                const uint32_t uy0 = (uint32_t)i0y, uy1 = (uint32_t)i1y;
                const uint32_t ux0 = (uint32_t)i0x, ux1 = (uint32_t)i1x;

                const uint32_t lo0 = uy0 * PS1_32;
                const uint32_t hi0 = __umulhi(uy0, PS1_32);
                const uint32_t L0  = lo0 & 0xFFFFFu;
                const uint32_t H0  = (lo0 >> 20) | (hi0 << 12);      // < 2^24
                const uint32_t b0  = N_ENTRIES_C - (42u * H0) % N_ENTRIES_C;

                const uint32_t lo1 = uy1 * PS1_32;
                const uint32_t hi1 = __umulhi(uy1, PS1_32);
                const uint32_t L1  = lo1 & 0xFFFFFu;
                const uint32_t H1  = (lo1 >> 20) | (hi1 << 12);
                const uint32_t b1  = N_ENTRIES_C - (42u * H1) % N_ENTRIES_C;

                const float2* __restrict__ dbase = data + off[l];
                v00 = dbase[hash_fold(L0, b0, ux0)];
                v01 = dbase[hash_fold(L1, b1, ux0)];
                v10 = dbase[hash_fold(L0, b0, ux1)];
                v11 = dbase[hash_fold(L1, b1, ux1)];
            }

            const float r0  = w00 * v00.x + w01 * v01.x + w10 * v10.x + w11 * v11.x;
            const float r1v = w00 * v00.y + w01 * v01.y + w10 * v10.y + w11 * v11.y;
            op[l] = make_float2(r0, r1v);
        }
    }
}

extern "C" void kernel_launch(void* const* d_in, const int* in_sizes, int n_in,
                              void* d_out, int out_size, void* d_ws, size_t ws_size,
                              hipStream_t stream) {
    const float2* x    = (const float2*)d_in[0];   // [N_POINTS, 2] f32
    const float2* data = (const float2*)d_in[1];   // [TABLE_SIZE, 2] f32
    float* out = (float*)d_out;                    // [N_POINTS, 32] f32

    // 1024 blocks x 256 threads: 2 points/thread grid-stride; 44.8 KB LDS/WG
    // allows ~7 concurrent workgroups per WGP (320 KB LDS) for latency hiding.
    hashgrid_encode_kernel<<<1024, 256, 0, stream>>>(x, data, out);
}